// FMSpectralConv2D_38405597561167
// MI455X (gfx1250) — compile-verified
//
#include <hip/hip_runtime.h>
#include <hip/hip_bf16.h>
#include <cmath>

// ---------------------------------------------------------------------------
// FNO spectral conv on MI455X (gfx1250): 4 truncated-DFT GEMMs on
// V_WMMA_F32_16X16X4_F32 (wave32), with Tensor-Data-Mover (TDM) async
// HBM->LDS staging double-buffered against WMMA compute.
// ---------------------------------------------------------------------------

typedef float v2f __attribute__((ext_vector_type(2)));
typedef float v8f __attribute__((ext_vector_type(8)));
typedef unsigned int v4u __attribute__((ext_vector_type(4)));
typedef int v4i __attribute__((ext_vector_type(4)));
typedef int v8i __attribute__((ext_vector_type(8)));

#if defined(__gfx1250__) && __has_builtin(__builtin_amdgcn_wmma_f32_16x16x4_f32)
#define HAVE_WMMA_F32X4 1
#endif

#if defined(__gfx1250__) && __has_builtin(__builtin_amdgcn_tensor_load_to_lds)
#define HAVE_TDM 1
#endif

__device__ __forceinline__ v8f wmma_f32(v2f a, v2f b, v8f c) {
#ifdef HAVE_WMMA_F32X4
  // 8 args: (neg_a, A, neg_b, B, c_mod, C, reuse_a, reuse_b)
  return __builtin_amdgcn_wmma_f32_16x16x4_f32(false, a, false, b, (short)0, c,
                                               false, false);
#else
  c[0] += a.x * b.x + a.y * b.y;  // compile-probe fallback only
  return c;
#endif
}

#ifdef HAVE_TDM
// Issue a 1-D TENSOR_LOAD_TO_LDS: nelems f32 from gaddr -> LDS byte offset.
// D# per CDNA5 ISA. Tracked by TENSORcnt.
__device__ __forceinline__ void tdm_load_1d(unsigned lds_off, const void* gaddr,
                                            unsigned nelems) {
  unsigned long long ga = (unsigned long long)(size_t)gaddr;
  v4u g0;
  g0[0] = 1u;                                   // count=1, user descriptor
  g0[1] = lds_off;                              // lds_addr (bytes)
  g0[2] = (unsigned)(ga & 0xffffffffu);         // global_addr[31:0]
  g0[3] = (unsigned)((ga >> 32) & 0x1ffffffu)   // global_addr[56:32]
          | (2u << 30);                         // type=2 ("image")
  v8i g1 = {};
  g1[0] = (int)(2u << 16);                      // data_size=4B, no flags
  g1[1] = (int)((nelems & 0xffffu) << 16);      // tensor_dim0[15:0]
  g1[2] = (int)(((nelems >> 16) & 0xffffu)      // tensor_dim0[31:16]
                | (1u << 16));                  // tensor_dim1 = 1
  g1[3] = (int)((nelems & 0xffffu) << 16);      // tile_dim0 = nelems (1-D tile)
  v4i g2 = {};                                  // dims 2/3 unused
  v4i g3 = {};
#if __clang_major__ >= 23
  v8i g4 = {};
  __builtin_amdgcn_tensor_load_to_lds(g0, g1, g2, g3, g4, 0);
#else
  __builtin_amdgcn_tensor_load_to_lds(g0, g1, g2, g3, 0);
#endif
}
#endif

// Problem constants
#define HH 1024
#define WW 1024
#define CIN 32
#define COUT 32
#define NM0 64
#define NM1 33           // rfft modes along W
#define NRI 66           // 2*NM1 (Re block then Im block)
#define TEMB 256
#define NMODES 2112      // NM0*NM1

// Workspace layout (floats)
#define OFF_A1 0                         // 80 x 1024   (stage1 A, padded)
#define SZ_A1 (80 * 1024)
#define OFF_A2 (OFF_A1 + SZ_A1)          // 128 x 2048  (stage2 A)
#define SZ_A2 (128 * 2048)
#define OFF_A4 (OFF_A2 + SZ_A2)          // 1024 x 128  (stage4 A)
#define SZ_A4 (1024 * 128)
#define OFF_A5 (OFF_A4 + SZ_A4)          // 1024 x 68   (stage5 A, padded K)
#define SZ_A5 (1024 * 68)
#define OFF_PFT (OFF_A5 + SZ_A5)         // 2 x 2112    (phi_t_ft re/im)
#define SZ_PFT (2 * NMODES)
#define OFF_U (OFF_PFT + SZ_PFT)         // 1024 x 2112 == (2048 x 1056)
#define SZ_U (HH * NRI * CIN)
#define OFF_V (OFF_U + SZ_U)             // 128 x 1056  (v_r ; v_i)
#define SZ_V (128 * 1056)
#define OFF_B4 (OFF_V + SZ_V)            // 128 x 2112  ([yr;-yi | yi;yr])
#define SZ_B4 (128 * 2112)
#define OFF_T (OFF_B4 + SZ_B4)           // 1024 x 2112 (t_r | t_i)
#define SZ_T (HH * 2112)

// ---------------------------------------------------------------------------
// Twiddle tables; phases reduced exactly via (m*n) mod 1024.
// ---------------------------------------------------------------------------
__global__ void init_tables(float* __restrict__ A1, float* __restrict__ A2,
                            float* __restrict__ A4, float* __restrict__ A5) {
  const int idx = blockIdx.x * blockDim.x + threadIdx.x;
  const float th = 6.28318530717958647692f / 1024.f;
  const float NORM = 1.0f / (1024.0f * 1024.0f);  // rfft2 'forward' norm

  if (idx < SZ_A1) {  // S1: u[m1] = sum_w x[w] e^{-i th m1 w} * NORM
    int r = idx >> 10, w = idx & 1023;
    float v = 0.f;
    if (r < NM1) {
      int p = (r * w) & 1023;
      v = NORM * cosf(th * (float)p);
    } else if (r < NRI) {
      int m1 = r - NM1;
      int p = (m1 * w) & 1023;
      v = -NORM * sinf(th * (float)p);
    }
    A1[idx] = v;
  }
  if (idx < SZ_A2) {  // S2: v = sum_h (cos - i sin)(u_r + i u_i)
    int m = idx >> 11, t = idx & 2047;
    int hh = t >> 1, part = t & 1;
    int m0 = m & 63;
    int p = (m0 * hh) & 1023;
    float ang = th * (float)p;
    float c = cosf(ang), s = sinf(ang);
    A2[idx] = (m < NM0) ? (part ? s : c)      // v_r row: [c, s]
                        : (part ? c : -s);    // v_i row: [-s, c]
  }
  if (idx < SZ_A4) {  // S4: t = sum_m0 y e^{+i th m0 h}
    int hh = idx >> 7, k = idx & 127;
    int m = k & 63;
    int p = (hh * m) & 1023;
    float ang = th * (float)p;
    A4[idx] = (k < NM0) ? cosf(ang) : sinf(ang);
  }
  if (idx < SZ_A5) {  // S5: out = sum_m1 c (t_r cos - t_i sin), c0=1 else 2
    int w = idx / 68, q = idx % 68;
    float v = 0.f;
    if (q < NRI) {
      int m1 = (q < NM1) ? q : q - NM1;
      float cf = (m1 == 0) ? 1.f : 2.f;
      int p = (w * m1) & 1023;
      float ang = th * (float)p;
      v = (q < NM1) ? cf * cosf(ang) : -cf * sinf(ang);
    }
    A5[idx] = v;
  }
}

// phi_t_ft[j] = phi_t @ (W_real + i W_imag), j = m0*33 + m1
__global__ void phi_kernel(const float* __restrict__ phi,
                           const float* __restrict__ Wr,
                           const float* __restrict__ Wi,
                           float* __restrict__ pft) {
  int j = blockIdx.x * blockDim.x + threadIdx.x;
  if (j >= NMODES) return;
  float ar = 0.f, ai = 0.f;
  for (int t = 0; t < TEMB; ++t) {
    float p = phi[t];
    ar += p * Wr[t * NMODES + j];
    ai += p * Wi[t * NMODES + j];
  }
  pft[j] = ar;
  pft[NMODES + j] = ai;
}

// ---------------------------------------------------------------------------
// Stage 1: per-row forward DFT along W. Per h: C(80x32) = A1(80x1024) x x[h].
// x row streamed HBM->LDS in 16KB chunks via TDM, double-buffered against
// WMMA compute (wave 0 drives the DMA; TENSORcnt + barrier sync).
// U row layout: [Re(m1,k) 1056 | Im(m1,k) 1056] -> doubles as the (2048x1056)
// B matrix of stage 2.
// ---------------------------------------------------------------------------
__global__ __launch_bounds__(160) void stage1(const float* __restrict__ x,
                                              const float* __restrict__ A1,
                                              float* __restrict__ U) {
  __shared__ float lds[2][4096];  // 2 x 16KB chunk of one x row
  const int tid = threadIdx.x;
  const int lane = tid & 31, wv = tid >> 5;  // wv = M-tile 0..4
  const int lm = lane & 15, lh = lane >> 4;
  const int h = blockIdx.x;
  const float* xrow = x + (size_t)h * (WW * CIN);
  const float* Abase = A1 + (wv * 16 + lm) * 1024;
  v8f acc0 = {}, acc1 = {};

#ifdef HAVE_TDM
  const unsigned ldsbase = (unsigned)(size_t)(&lds[0][0]);
  if (wv == 0) tdm_load_1d(ldsbase, xrow, 4096);
#endif

  for (int c = 0; c < 8; ++c) {  // 8 chunks of 128 w-columns
#ifdef HAVE_TDM
    if (wv == 0) __builtin_amdgcn_s_wait_tensorcnt(0);  // chunk c landed
    __syncthreads();
    if (wv == 0 && c + 1 < 8)  // prefetch next chunk while we compute
      tdm_load_1d(ldsbase + (unsigned)(((c + 1) & 1) * 16384),
                  xrow + (c + 1) * 4096, 4096);
    const float* buf = lds[c & 1];
#else
    __syncthreads();
    for (int i = tid; i < 1024; i += 160)
      ((float4*)lds[0])[i] = ((const float4*)(xrow + c * 4096))[i];
    __syncthreads();
    const float* buf = lds[0];
#endif
    const float* Ac = Abase + c * 128;
    for (int kb = 0; kb < 128; kb += 4) {
      const int kr = kb + 2 * lh;                 // A/B: K = 2*(lane>>4)+j
      v2f a = *(const v2f*)(Ac + kr);
      v2f b0, b1;
      b0.x = buf[kr * CIN + lm];
      b0.y = buf[(kr + 1) * CIN + lm];
      b1.x = buf[kr * CIN + 16 + lm];
      b1.y = buf[(kr + 1) * CIN + 16 + lm];
      acc0 = wmma_f32(a, b0, acc0);
      acc1 = wmma_f32(a, b1, acc1);
    }
  }
  float* Urow = U + (size_t)h * (NRI * CIN);
#pragma unroll
  for (int r = 0; r < 8; ++r) {
    int m = wv * 16 + r + 8 * lh;  // C/D: M = r + 8*(lane>>4)
    if (m < NRI) {
      Urow[m * CIN + lm] = acc0[r];
      Urow[m * CIN + 16 + lm] = acc1[r];
    }
  }
}

// ---------------------------------------------------------------------------
// Stage 2: V(128x1056) = A2(128x2048) x U2(2048x1056). 66 blocks of 4 waves,
// each wave owns 2 M-tiles for one N-tile, full K sweep. A2 is L2-resident.
// ---------------------------------------------------------------------------
__global__ __launch_bounds__(128) void stage2(const float* __restrict__ A2,
                                              const float* __restrict__ U,
                                              float* __restrict__ V) {
  const int nt = blockIdx.x;  // N-tile 0..65
  const int tid = threadIdx.x;
  const int lane = tid & 31, wv = tid >> 5;
  const int lm = lane & 15, lh = lane >> 4;
  const int mt0 = wv * 2;
  const float* Ub = U + nt * 16 + lm;
  const float* A0 = A2 + (mt0 * 16 + lm) * 2048;
  const float* A1p = A2 + ((mt0 + 1) * 16 + lm) * 2048;
  v8f acc0 = {}, acc1 = {};
  for (int kb = 0; kb < 2048; kb += 4) {
    const int kr = kb + 2 * lh;
    v2f b;
    b.x = Ub[(size_t)kr * 1056];
    b.y = Ub[(size_t)(kr + 1) * 1056];
    v2f a0 = *(const v2f*)(A0 + kr);
    v2f a1 = *(const v2f*)(A1p + kr);
    acc0 = wmma_f32(a0, b, acc0);
    acc1 = wmma_f32(a1, b, acc1);
  }
#pragma unroll
  for (int r = 0; r < 8; ++r) {
    const int n = nt * 16 + lm;
    V[(size_t)(mt0 * 16 + r + 8 * lh) * 1056 + n] = acc0[r];
    V[(size_t)((mt0 + 1) * 16 + r + 8 * lh) * 1056 + n] = acc1[r];
  }
}

// ---------------------------------------------------------------------------
// Stage 3: per-mode 32x32 complex channel mix + phi modulation. Writes the
// duplicated B4(128x2112) = [[y_r | y_i], [-y_i | y_r]] so stage 4 is ONE GEMM.
// ---------------------------------------------------------------------------
__global__ __launch_bounds__(32) void modemix(const float* __restrict__ V,
                                              const float* __restrict__ wr,
                                              const float* __restrict__ wi,
                                              const float* __restrict__ pft,
                                              float* __restrict__ B4) {
  const int j = blockIdx.x;  // 0..2111 = m0*33+m1
  const int m0 = j / NM1, m1 = j % NM1;
  const int l = threadIdx.x;  // out channel
  const float* vr = V + (size_t)m0 * 1056 + m1 * 32;
  const float* vi = V + (size_t)(NM0 + m0) * 1056 + m1 * 32;
  const float* wrp = wr + (size_t)j * (CIN * COUT);
  const float* wip = wi + (size_t)j * (CIN * COUT);
  float yr = 0.f, yi = 0.f;
  for (int k = 0; k < CIN; ++k) {
    float a = vr[k], b = vi[k];
    float c = wrp[k * COUT + l], d = wip[k * COUT + l];
    yr += a * c - b * d;
    yi += a * d + b * c;
  }
  const float pr = pft[j], pi = pft[NMODES + j];
  const float zr = pr * yr - pi * yi;
  const float zi = pr * yi + pi * yr;
  const int n = m1 * 32 + l;
  B4[(size_t)m0 * 2112 + n] = zr;
  B4[(size_t)(NM0 + m0) * 2112 + n] = -zi;
  B4[(size_t)m0 * 2112 + 1056 + n] = zi;
  B4[(size_t)(NM0 + m0) * 2112 + 1056 + n] = zr;
}

// ---------------------------------------------------------------------------
// Stage 4: T(1024x2112) = A4(1024x128) x B4(128x2112). Grid (64 M-tiles, 33
// N-groups), 4 waves per block (one N-tile each), K=128 -> 32 WMMA steps.
// T row h = [t_r(m1,l) | t_i(m1,l)].
// ---------------------------------------------------------------------------
__global__ __launch_bounds__(128) void stage4(const float* __restrict__ A4,
                                              const float* __restrict__ B4,
                                              float* __restrict__ T) {
  const int mt = blockIdx.x;
  const int tid = threadIdx.x;
  const int lane = tid & 31, wv = tid >> 5;
  const int nt = blockIdx.y * 4 + wv;  // 0..131
  const int lm = lane & 15, lh = lane >> 4;
  const float* Ab = A4 + (mt * 16 + lm) * 128;
  const float* Bb = B4 + nt * 16 + lm;
  v8f acc = {};
  for (int kb = 0; kb < 128; kb += 4) {
    const int kr = kb + 2 * lh;
    v2f a = *(const v2f*)(Ab + kr);
    v2f b;
    b.x = Bb[(size_t)kr * 2112];
    b.y = Bb[(size_t)(kr + 1) * 2112];
    acc = wmma_f32(a, b, acc);
  }
#pragma unroll
  for (int r = 0; r < 8; ++r)
    T[(size_t)(mt * 16 + r + 8 * lh) * 2112 + nt * 16 + lm] = acc[r];
}

// ---------------------------------------------------------------------------
// Stage 5: per-row inverse along W with irfft symmetry factors baked into A5.
// Per h: out[w,l] = A5(1024x68) x T_row(68x32, zero-padded K in LDS).
// T row staged via one TDM load, shared by 8 waves; output lands directly in
// final (h,w,c) layout.
// ---------------------------------------------------------------------------
__global__ __launch_bounds__(256) void stage5(const float* __restrict__ A5,
                                              const float* __restrict__ T,
                                              float* __restrict__ out) {
  __shared__ float bl[68 * 32];
  const int h = blockIdx.x;
  const int tid = threadIdx.x;
  const int lane = tid & 31, wv = tid >> 5;
#ifdef HAVE_TDM
  if (wv == 0)
    tdm_load_1d((unsigned)(size_t)(&bl[0]), T + (size_t)h * 2112, 2112);
  for (int i = 2112 + tid; i < 2176; i += 256) bl[i] = 0.f;  // K padding
  if (wv == 0) __builtin_amdgcn_s_wait_tensorcnt(0);
  __syncthreads();
#else
  for (int i = tid; i < 2176; i += 256)
    bl[i] = (i < 2112) ? T[(size_t)h * 2112 + i] : 0.f;
  __syncthreads();
#endif
  const int lm = lane & 15, lh = lane >> 4;
  float* orow = out + (size_t)h * (WW * COUT);
  for (int mt = wv; mt < 64; mt += 8) {
    const float* Ab = A5 + (mt * 16 + lm) * 68;
    v8f acc0 = {}, acc1 = {};
    for (int kb = 0; kb < 68; kb += 4) {
      const int kr = kb + 2 * lh;
      v2f a = *(const v2f*)(Ab + kr);
      v2f b0, b1;
      b0.x = bl[kr * 32 + lm];
      b0.y = bl[(kr + 1) * 32 + lm];
      b1.x = bl[kr * 32 + 16 + lm];
      b1.y = bl[(kr + 1) * 32 + 16 + lm];
      acc0 = wmma_f32(a, b0, acc0);
      acc1 = wmma_f32(a, b1, acc1);
    }
#pragma unroll
    for (int r = 0; r < 8; ++r) {
      const int w = mt * 16 + r + 8 * lh;
      orow[w * COUT + lm] = acc0[r];
      orow[w * COUT + 16 + lm] = acc1[r];
    }
  }
}

// ---------------------------------------------------------------------------
extern "C" void kernel_launch(void* const* d_in, const int* in_sizes, int n_in,
                              void* d_out, int out_size, void* d_ws,
                              size_t ws_size, hipStream_t stream) {
  (void)in_sizes; (void)n_in; (void)out_size; (void)ws_size;
  const float* x = (const float*)d_in[0];
  const float* phi = (const float*)d_in[1];
  const float* wr = (const float*)d_in[2];
  const float* wi = (const float*)d_in[3];
  const float* Wre = (const float*)d_in[4];
  const float* Wim = (const float*)d_in[5];
  float* out = (float*)d_out;
  float* ws = (float*)d_ws;

  float* A1 = ws + OFF_A1;
  float* A2 = ws + OFF_A2;
  float* A4 = ws + OFF_A4;
  float* A5 = ws + OFF_A5;
  float* PFT = ws + OFF_PFT;
  float* U = ws + OFF_U;
  float* V = ws + OFF_V;
  float* B4 = ws + OFF_B4;
  float* T = ws + OFF_T;

  init_tables<<<(SZ_A2 + 255) / 256, 256, 0, stream>>>(A1, A2, A4, A5);
  phi_kernel<<<(NMODES + 255) / 256, 256, 0, stream>>>(phi, Wre, Wim, PFT);
  stage1<<<HH, 160, 0, stream>>>(x, A1, U);
  stage2<<<66, 128, 0, stream>>>(A2, U, V);
  modemix<<<NMODES, 32, 0, stream>>>(V, wr, wi, PFT, B4);
  stage4<<<dim3(64, 33), 128, 0, stream>>>(A4, B4, T);
  stage5<<<HH, 256, 0, stream>>>(A5, T, out);
}